// SRHGNLayer_41652592837288
// MI455X (gfx1250) — compile-verified
//
#include <hip/hip_runtime.h>
#include <math.h>

#define N_NODES 50000
#define N_EDGES 600000
#define DIM     128
#define NH      4      // node heads
#define NTH     4      // type heads
#define ALPHA   0.5f
#define EPSN    1e-9f

typedef __attribute__((ext_vector_type(2))) float v2f;
typedef __attribute__((ext_vector_type(8))) float v8f;

// ---------- helpers ----------
__device__ __forceinline__ float leaky(float x) { return x > 0.0f ? x : 0.01f * x; }

// monotonic uint key for float atomic max
__device__ __forceinline__ unsigned fkey(float f) {
  unsigned u = __float_as_uint(f);
  return (u & 0x80000000u) ? ~u : (u | 0x80000000u);
}
__device__ __forceinline__ float fdekey(unsigned k) {
  return __uint_as_float((k & 0x80000000u) ? (k & 0x7FFFFFFFu) : ~k);
}

__device__ __forceinline__ float wave_allred(float v) {
  // wave32 xor-tree reduction
  for (int m = 16; m >= 1; m >>= 1) v += __shfl_xor(v, m, 32);
  return v;
}

// ---------- fills ----------
__global__ void fill_f32(float* __restrict__ p, float v, int n) {
  int i = blockIdx.x * blockDim.x + threadIdx.x;
  if (i < n) p[i] = v;
}
__global__ void fill_u32(unsigned* __restrict__ p, unsigned v, int n) {
  int i = blockIdx.x * blockDim.x + threadIdx.x;
  if (i < n) p[i] = v;
}

// ---------- f32 WMMA GEMM:  C[N,128] = A[N,128] @ W[128,128] + b ----------
// block = 256 threads = 8 waves; blockIdx.x = 16-row tile; wave w = 16-col tile w.
// V_WMMA_F32_16X16X4_F32: A frag (16x4) = v2f: lane m=l%16, K = 2*(l/16)+{0,1}
//                         B frag (4x16) = v2f: lane n=l%16, K = 2*(l/16)+{0,1}
//                         C/D = v8f: M = vgpr + 8*(l/16), N = l%16
__global__ __launch_bounds__(256) void gemm_wmma_f32(
    const float* __restrict__ A, const float* __restrict__ W,
    const float* __restrict__ b, float* __restrict__ C) {
  const int wave = threadIdx.x >> 5;
  const int lane = threadIdx.x & 31;
  const int half = lane >> 4;   // 0 or 1
  const int l16  = lane & 15;
  const int row0 = blockIdx.x * 16;
  const int col0 = wave * 16;

  v8f acc = {};
  const float* arow = A + (size_t)(row0 + l16) * DIM;
#pragma unroll 4
  for (int k0 = 0; k0 < DIM; k0 += 4) {
    v2f af, bf;
    const float* ap = arow + k0 + half * 2;
    af.x = ap[0];
    af.y = ap[1];
    const float* wp = W + (size_t)(k0 + half * 2) * DIM + col0 + l16;
    bf.x = wp[0];
    bf.y = wp[DIM];
    acc = __builtin_amdgcn_wmma_f32_16x16x4_f32(
        /*neg_a=*/false, af, /*neg_b=*/false, bf,
        /*c_mod=*/(short)0, acc, /*reuse_a=*/false, /*reuse_b=*/false);
  }

  const float bias = b[col0 + l16];
  float* crow = C + (size_t)(row0 + half * 8) * DIM + col0 + l16;
#pragma unroll
  for (int i = 0; i < 8; ++i) crow[(size_t)i * DIM] = acc[i] + bias;
}

// ---------- tiny projection: out[N,4] = X[N,128] @ W[128,4] + b ----------
__global__ void proj4_kernel(const float* __restrict__ X, const float* __restrict__ W,
                             const float* __restrict__ b, float* __restrict__ out) {
  int i = blockIdx.x * blockDim.x + threadIdx.x;
  if (i >= N_NODES * NH) return;
  int node = i >> 2, hh = i & 3;
  float acc = b[hh];
  const float* x = X + (size_t)node * DIM;
#pragma unroll 8
  for (int d = 0; d < DIM; ++d) acc += x[d] * W[d * NH + hh];
  out[i] = acc;
}

// ---------- edge softmax phase 1: logits + segment max ----------
__global__ void edge_logit_max_kernel(const int* __restrict__ src, const int* __restrict__ dst,
                                      const float* __restrict__ asrc, const float* __restrict__ adst,
                                      float* __restrict__ elog, unsigned* __restrict__ umax) {
  int i = blockIdx.x * blockDim.x + threadIdx.x;
  if (i >= N_EDGES * NH) return;
  int e = i >> 2, hh = i & 3;
  int s = src[e], d = dst[e];
  float t = leaky(asrc[s * NH + hh] + adst[d * NH + hh]);
  elog[i] = t;
  atomicMax(&umax[d * NH + hh], fkey(t));
}

// ---------- edge softmax phase 2: exp + segment sum ----------
__global__ void edge_exp_sum_kernel(const int* __restrict__ dst, float* __restrict__ elog,
                                    const unsigned* __restrict__ umax, float* __restrict__ ssum) {
  int i = blockIdx.x * blockDim.x + threadIdx.x;
  if (i >= N_EDGES * NH) return;
  int e = i >> 2, hh = i & 3;
  int d = dst[e];
  float ea = expf(elog[i] - fdekey(umax[d * NH + hh]));
  elog[i] = ea;
  atomicAdd(&ssum[d * NH + hh], ea);
}

// ---------- message scatter: z[dst] += hs[src] * attn ----------
// 32 threads per edge, 4 floats each (float4 gather + 4 global_atomic_add_f32)
__global__ void edge_scatter_kernel(const int* __restrict__ src, const int* __restrict__ dst,
                                    const float* __restrict__ elog, const float* __restrict__ ssum,
                                    const float* __restrict__ hs, float* __restrict__ z) {
  int i = blockIdx.x * blockDim.x + threadIdx.x;
  if (i >= N_EDGES * 32) return;
  int e = i >> 5, q = i & 31;
  int head = q >> 3;          // (q*4)/32
  int d0 = q * 4;
  int s = src[e], d = dst[e];
  float att = elog[e * NH + head] / ssum[d * NH + head];
  const float4 v = *(const float4*)(hs + (size_t)s * DIM + d0);
  float* zp = z + (size_t)d * DIM + d0;
  atomicAdd(zp + 0, v.x * att);
  atomicAdd(zp + 1, v.y * att);
  atomicAdd(zp + 2, v.z * att);
  atomicAdd(zp + 3, v.w * att);
}

// ---------- relation attention (tiny, 8 values total) ----------
__global__ void rel_attn_kernel(const float* __restrict__ rel_emb,
                                const float* __restrict__ Wr0, const float* __restrict__ br0,
                                const float* __restrict__ Wr1, const float* __restrict__ br1,
                                float* __restrict__ relb) {
  int hh = threadIdx.x;
  if (hh >= NTH) return;
  float r0 = br0[hh], r1 = br1[hh];
  for (int d = 0; d < DIM; ++d) {
    r0 += rel_emb[d] * Wr0[d * NTH + hh];
    r1 += rel_emb[DIM + d] * Wr1[d * NTH + hh];
  }
  r0 = leaky(r0);
  r1 = leaky(r1);
  float m = fmaxf(r0, r1);
  float e0 = expf(r0 - m), e1 = expf(r1 - m);
  float inv = 1.0f / (e0 + e1);
  relb[hh] = e0 * inv;
  relb[NTH + hh] = e1 * inv;
}

// ---------- semantic attention + final attn outputs (wave per node) ----------
__global__ __launch_bounds__(256) void sem_attn_kernel(
    const float* __restrict__ z0, const float* __restrict__ z1, const float* __restrict__ hd,
    const float* __restrict__ Ws0, const float* __restrict__ bs0,
    const float* __restrict__ Wd0, const float* __restrict__ bd0,
    const float* __restrict__ Ws1, const float* __restrict__ bs1,
    const float* __restrict__ Wd1, const float* __restrict__ bd1,
    const float* __restrict__ relb,
    float* __restrict__ attn_out, float* __restrict__ sem_out, float* __restrict__ rel_out) {
  int node = blockIdx.x * 8 + (threadIdx.x >> 5);
  if (node >= N_NODES) return;
  int lane = threadIdx.x & 31;
  size_t base = (size_t)node * DIM + lane * 4;

  float a0[4], a1[4], ad[4];
  {
    float4 t0 = *(const float4*)(z0 + base);
    float4 t1 = *(const float4*)(z1 + base);
    float4 td = *(const float4*)(hd + base);
    a0[0] = t0.x; a0[1] = t0.y; a0[2] = t0.z; a0[3] = t0.w;
    a1[0] = t1.x; a1[1] = t1.y; a1[2] = t1.z; a1[3] = t1.w;
    ad[0] = td.x; ad[1] = td.y; ad[2] = td.z; ad[3] = td.w;
  }
  float ss0 = 0, ss1 = 0, ssd = 0;
#pragma unroll
  for (int j = 0; j < 4; ++j) { ss0 += a0[j]*a0[j]; ss1 += a1[j]*a1[j]; ssd += ad[j]*ad[j]; }
  float inv0 = 1.0f / fmaxf(sqrtf(wave_allred(ss0)), EPSN);
  float inv1 = 1.0f / fmaxf(sqrtf(wave_allred(ss1)), EPSN);
  float invd = 1.0f / fmaxf(sqrtf(wave_allred(ssd)), EPSN);

  float s0[NTH] = {0, 0, 0, 0}, s1[NTH] = {0, 0, 0, 0};
#pragma unroll
  for (int j = 0; j < 4; ++j) {
    int d = lane * 4 + j;
    float u0 = a0[j] * inv0, u1 = a1[j] * inv1, ud = ad[j] * invd;
#pragma unroll
    for (int hh = 0; hh < NTH; ++hh) {
      s0[hh] += u0 * Ws0[d * NTH + hh] + ud * Wd0[d * NTH + hh];
      s1[hh] += u1 * Ws1[d * NTH + hh] + ud * Wd1[d * NTH + hh];
    }
  }
#pragma unroll
  for (int hh = 0; hh < NTH; ++hh) { s0[hh] = wave_allred(s0[hh]); s1[hh] = wave_allred(s1[hh]); }

  if (lane == 0) {
    size_t o = (size_t)node * 2 * NTH;
#pragma unroll
    for (int hh = 0; hh < NTH; ++hh) {
      float l0 = leaky(s0[hh] + bs0[hh] + bd0[hh]);
      float l1 = leaky(s1[hh] + bs1[hh] + bd1[hh]);
      float m = fmaxf(l0, l1);
      float e0 = expf(l0 - m), e1 = expf(l1 - m);
      float inv = 1.0f / (e0 + e1);
      float p0 = e0 * inv, p1 = e1 * inv;
      float r0 = relb[hh], r1 = relb[NTH + hh];
      sem_out[o + hh] = p0;        sem_out[o + NTH + hh] = p1;
      rel_out[o + hh] = r0;        rel_out[o + NTH + hh] = r1;
      attn_out[o + hh] = ALPHA * p0 + (1.0f - ALPHA) * r0;
      attn_out[o + NTH + hh] = ALPHA * p1 + (1.0f - ALPHA) * r1;
    }
  }
}

// ---------- final: z_out = normalize(gelu(z0 + z1 + h)) (wave per node) ----------
__global__ __launch_bounds__(256) void final_out_kernel(
    const float* __restrict__ z0, const float* __restrict__ z1,
    const float* __restrict__ h, float* __restrict__ out) {
  int node = blockIdx.x * 8 + (threadIdx.x >> 5);
  if (node >= N_NODES) return;
  int lane = threadIdx.x & 31;
  size_t base = (size_t)node * DIM + lane * 4;
  float g[4];
  float ss = 0.0f;
#pragma unroll
  for (int j = 0; j < 4; ++j) {
    float x = z0[base + j] + z1[base + j] + h[base + j];
    float gg = 0.5f * x * (1.0f + erff(x * 0.70710678118654752f)); // exact gelu
    g[j] = gg;
    ss += gg * gg;
  }
  float inv = 1.0f / fmaxf(sqrtf(wave_allred(ss)), EPSN);
#pragma unroll
  for (int j = 0; j < 4; ++j) out[base + j] = g[j] * inv;
}

// ---------- launch ----------
extern "C" void kernel_launch(void* const* d_in, const int* in_sizes, int n_in,
                              void* d_out, int out_size, void* d_ws, size_t ws_size,
                              hipStream_t stream) {
  const float* h      = (const float*)d_in[0];
  const int* src[2]   = {(const int*)d_in[1], (const int*)d_in[3]};
  const int* dst[2]   = {(const int*)d_in[2], (const int*)d_in[4]};
  const float* W_node = (const float*)d_in[5];
  const float* b_node = (const float*)d_in[6];
  const float* W_e[2] = {(const float*)d_in[7], (const float*)d_in[9]};
  const float* b_e[2] = {(const float*)d_in[8], (const float*)d_in[10]};
  const float* Wsrc = (const float*)d_in[11], *bsrc = (const float*)d_in[12];
  const float* Wdst = (const float*)d_in[13], *bdst = (const float*)d_in[14];
  const float* Wsem_s[2] = {(const float*)d_in[15], (const float*)d_in[19]};
  const float* bsem_s[2] = {(const float*)d_in[16], (const float*)d_in[20]};
  const float* Wsem_d[2] = {(const float*)d_in[17], (const float*)d_in[21]};
  const float* bsem_d[2] = {(const float*)d_in[18], (const float*)d_in[22]};
  const float* Wrel0 = (const float*)d_in[23], *brel0 = (const float*)d_in[24];
  const float* Wrel1 = (const float*)d_in[25], *brel1 = (const float*)d_in[26];
  const float* rel_emb = (const float*)d_in[27];

  // workspace layout (floats)
  float* ws   = (float*)d_ws;
  float* hd   = ws;                                   // N*D
  float* hs   = hd + (size_t)N_NODES * DIM;           // N*D (reused per relation)
  float* z0   = hs + (size_t)N_NODES * DIM;           // N*D
  float* z1   = z0 + (size_t)N_NODES * DIM;           // N*D
  float* adst = z1 + (size_t)N_NODES * DIM;           // N*NH
  float* asrc = adst + (size_t)N_NODES * NH;          // N*NH
  unsigned* umax = (unsigned*)(asrc + (size_t)N_NODES * NH); // N*NH
  float* ssum = (float*)umax + (size_t)N_NODES * NH;  // N*NH
  float* elog = ssum + (size_t)N_NODES * NH;          // E*NH
  float* relb = elog + (size_t)N_EDGES * NH;          // 8
  float* zrel[2] = {z0, z1};

  // output layout
  float* out      = (float*)d_out;
  float* z_out    = out;                                   // N*D
  float* attn_out = out + (size_t)N_NODES * DIM;           // N*2*NTH
  float* sem_out  = attn_out + (size_t)N_NODES * 2 * NTH;  // N*2*NTH
  float* rel_out  = sem_out + (size_t)N_NODES * 2 * NTH;   // N*2*NTH

  const int TPB = 256;
  const unsigned NEG_INF_KEY = 0x007FFFFFu; // fkey(-inf)

  // zero accumulators (every call; deterministic)
  {
    int n = 2 * N_NODES * DIM;
    fill_f32<<<(n + TPB - 1) / TPB, TPB, 0, stream>>>(z0, 0.0f, n);
  }

  // hd = h @ W_node + b_node   (f32 WMMA)
  gemm_wmma_f32<<<N_NODES / 16, TPB, 0, stream>>>(h, W_node, b_node, hd);
  // adst = hd @ Wdst + bdst
  proj4_kernel<<<(N_NODES * NH + TPB - 1) / TPB, TPB, 0, stream>>>(hd, Wdst, bdst, adst);
  // relation-level attention (broadcast values)
  rel_attn_kernel<<<1, 64, 0, stream>>>(rel_emb, Wrel0, brel0, Wrel1, brel1, relb);

  for (int r = 0; r < 2; ++r) {
    // hs = h @ W_e[r] + b_e[r]   (f32 WMMA)
    gemm_wmma_f32<<<N_NODES / 16, TPB, 0, stream>>>(h, W_e[r], b_e[r], hs);
    // asrc = hs @ Wsrc + bsrc
    proj4_kernel<<<(N_NODES * NH + TPB - 1) / TPB, TPB, 0, stream>>>(hs, Wsrc, bsrc, asrc);
    // reset segment max/sum
    {
      int n = N_NODES * NH;
      fill_u32<<<(n + TPB - 1) / TPB, TPB, 0, stream>>>(umax, NEG_INF_KEY, n);
      fill_f32<<<(n + TPB - 1) / TPB, TPB, 0, stream>>>(ssum, 0.0f, n);
    }
    int ne = N_EDGES * NH;
    edge_logit_max_kernel<<<(ne + TPB - 1) / TPB, TPB, 0, stream>>>(
        src[r], dst[r], asrc, adst, elog, umax);
    edge_exp_sum_kernel<<<(ne + TPB - 1) / TPB, TPB, 0, stream>>>(dst[r], elog, umax, ssum);
    int ns = N_EDGES * 32;
    edge_scatter_kernel<<<(ns + TPB - 1) / TPB, TPB, 0, stream>>>(
        src[r], dst[r], elog, ssum, hs, zrel[r]);
  }

  sem_attn_kernel<<<(N_NODES + 7) / 8, TPB, 0, stream>>>(
      z0, z1, hd,
      Wsem_s[0], bsem_s[0], Wsem_d[0], bsem_d[0],
      Wsem_s[1], bsem_s[1], Wsem_d[1], bsem_d[1],
      relb, attn_out, sem_out, rel_out);

  final_out_kernel<<<(N_NODES + 7) / 8, TPB, 0, stream>>>(z0, z1, h, z_out);
}